// TransformerMemoryDNC_45724221833219
// MI455X (gfx1250) — compile-verified
//
#include <hip/hip_runtime.h>
#include <math.h>

// Model constants (from reference)
#define BB   16
#define LLEN 64
#define EE   512
#define HH   1024
#define MMEM 128
#define HD   128
#define NRR  4
#define OO   512
#define CINN 1024
#define TOTD 1536
#define NHEAD 4
#define HDIM  32

typedef float v2f __attribute__((ext_vector_type(2)));
typedef float v8f __attribute__((ext_vector_type(8)));

__device__ __forceinline__ float sigm_(float x) { return 1.f / (1.f + __expf(-x)); }

// ---------------------------------------------------------------------------
// WMMA f32 GEMM, 16x64 tile per wave (4 independent accumulator chains):
//   C[Mr,N] = act(A[Mr,K](lda) @ W[N,K](ldw)^T + bias [+ A2@W2^T + bias2])
// Mr%16==0, N%64==0, K%4==0. Tile index is wave-uniform so EXEC stays all-1s
// for v_wmma (ISA requirement). act: 0=none, 1=exact GELU, 2=sigmoid, 3=tanh
// ---------------------------------------------------------------------------
__global__ void gemm64_wmma(const float* __restrict__ A, int lda,
                            const float* __restrict__ W, int ldw,
                            const float* __restrict__ bias,
                            float* __restrict__ C, int ldc,
                            int Mr, int N, int K, int act,
                            const float* __restrict__ A2, int lda2,
                            const float* __restrict__ W2, int ldw2, int K2,
                            const float* __restrict__ bias2)
{
    int lane = threadIdx.x & 31;
    int wave = threadIdx.x >> 5;
    int tilesN = N >> 6;                      // 64-wide tiles
    int tile = blockIdx.x * (blockDim.x >> 5) + wave;
    int totalTiles = (Mr >> 4) * tilesN;
    if (tile >= totalTiles) return;           // wave-uniform exit
    int tm  = (tile / tilesN) << 4;
    int tn0 = (tile % tilesN) << 6;

    int l    = lane & 15;                     // M (for A) / N (for B) index
    int koff = (lane >> 4) << 1;              // lanes 16-31 hold K+2,K+3

    v8f acc0 = {0.f,0.f,0.f,0.f,0.f,0.f,0.f,0.f};
    v8f acc1 = acc0, acc2 = acc0, acc3 = acc0;

    {
        const float* ap  = A + (size_t)(tm + l) * lda + koff;
        const float* wp0 = W + (size_t)(tn0 +  0 + l) * ldw + koff;
        const float* wp1 = W + (size_t)(tn0 + 16 + l) * ldw + koff;
        const float* wp2 = W + (size_t)(tn0 + 32 + l) * ldw + koff;
        const float* wp3 = W + (size_t)(tn0 + 48 + l) * ldw + koff;
        for (int k = 0; k < K; k += 4) {
            v2f a, b0, b1, b2, b3;
            a.x  = ap[k];  a.y  = ap[k + 1];
            b0.x = wp0[k]; b0.y = wp0[k + 1];
            b1.x = wp1[k]; b1.y = wp1[k + 1];
            b2.x = wp2[k]; b2.y = wp2[k + 1];
            b3.x = wp3[k]; b3.y = wp3[k + 1];
            acc0 = __builtin_amdgcn_wmma_f32_16x16x4_f32(false, a, false, b0, (short)0, acc0, false, false);
            acc1 = __builtin_amdgcn_wmma_f32_16x16x4_f32(false, a, false, b1, (short)0, acc1, false, false);
            acc2 = __builtin_amdgcn_wmma_f32_16x16x4_f32(false, a, false, b2, (short)0, acc2, false, false);
            acc3 = __builtin_amdgcn_wmma_f32_16x16x4_f32(false, a, false, b3, (short)0, acc3, false, false);
        }
    }
    if (A2) {   // fused second GEMM into same accumulators (LSTM gates)
        const float* ap  = A2 + (size_t)(tm + l) * lda2 + koff;
        const float* wp0 = W2 + (size_t)(tn0 +  0 + l) * ldw2 + koff;
        const float* wp1 = W2 + (size_t)(tn0 + 16 + l) * ldw2 + koff;
        const float* wp2 = W2 + (size_t)(tn0 + 32 + l) * ldw2 + koff;
        const float* wp3 = W2 + (size_t)(tn0 + 48 + l) * ldw2 + koff;
        for (int k = 0; k < K2; k += 4) {
            v2f a, b0, b1, b2, b3;
            a.x  = ap[k];  a.y  = ap[k + 1];
            b0.x = wp0[k]; b0.y = wp0[k + 1];
            b1.x = wp1[k]; b1.y = wp1[k + 1];
            b2.x = wp2[k]; b2.y = wp2[k + 1];
            b3.x = wp3[k]; b3.y = wp3[k + 1];
            acc0 = __builtin_amdgcn_wmma_f32_16x16x4_f32(false, a, false, b0, (short)0, acc0, false, false);
            acc1 = __builtin_amdgcn_wmma_f32_16x16x4_f32(false, a, false, b1, (short)0, acc1, false, false);
            acc2 = __builtin_amdgcn_wmma_f32_16x16x4_f32(false, a, false, b2, (short)0, acc2, false, false);
            acc3 = __builtin_amdgcn_wmma_f32_16x16x4_f32(false, a, false, b3, (short)0, acc3, false, false);
        }
    }

    // C/D layout: VGPR i, lanes 0-15 -> M=i, lanes 16-31 -> M=8+i; N = lane&15
    int mbase = tm + ((lane >> 4) << 3);
    v8f accs[4] = {acc0, acc1, acc2, acc3};
#pragma unroll
    for (int j = 0; j < 4; ++j) {
        int n = tn0 + 16 * j + l;
        float bi = (bias ? bias[n] : 0.f) + (bias2 ? bias2[n] : 0.f);
#pragma unroll
        for (int i = 0; i < 8; ++i) {
            float v = accs[j][i] + bi;
            if (act == 1)      v = 0.5f * v * (1.f + erff(v * 0.70710678118654752f));
            else if (act == 2) v = sigm_(v);
            else if (act == 3) v = tanhf(v);
            C[(size_t)(mbase + i) * ldc + n] = v;
        }
    }
}

// ---------------------------------------------------------------------------
// LayerNorm helpers (block of 256 threads; each thread owns a fixed j-stripe)
// ---------------------------------------------------------------------------
__device__ __forceinline__ void ln_finish(float* buf, float* red, int D,
                                          const float* __restrict__ g,
                                          const float* __restrict__ bv,
                                          float* __restrict__ out)
{
    int tid = threadIdx.x, nt = blockDim.x;
    float s = 0.f;
    for (int j = tid; j < D; j += nt) s += buf[j];
    red[tid] = s; __syncthreads();
    for (int off = nt >> 1; off > 0; off >>= 1) {
        if (tid < off) red[tid] += red[tid + off];
        __syncthreads();
    }
    float mu = red[0] / (float)D; __syncthreads();
    s = 0.f;
    for (int j = tid; j < D; j += nt) { float d = buf[j] - mu; s += d * d; }
    red[tid] = s; __syncthreads();
    for (int off = nt >> 1; off > 0; off >>= 1) {
        if (tid < off) red[tid] += red[tid + off];
        __syncthreads();
    }
    float rs = rsqrtf(red[0] / (float)D + 1e-5f);
    for (int j = tid; j < D; j += nt)
        out[j] = (buf[j] - mu) * rs * g[j] + bv[j];
}

__global__ void row_ln(const float* __restrict__ x, const float* __restrict__ res,
                       const float* __restrict__ g, const float* __restrict__ bv,
                       float* __restrict__ y, int D)
{
    __shared__ float buf[TOTD];
    __shared__ float red[256];
    int row = blockIdx.x;
    const float* xr = x + (size_t)row * D;
    for (int j = threadIdx.x; j < D; j += blockDim.x)
        buf[j] = xr[j] + (res ? res[(size_t)row * D + j] : 0.f);
    ln_finish(buf, red, D, g, bv, y + (size_t)row * D);
}

__global__ void input_ln(const float* __restrict__ x_emb, int t,
                         const float* __restrict__ rv,
                         const float* __restrict__ g, const float* __restrict__ bv,
                         float* __restrict__ inp)
{
    __shared__ float buf[CINN];
    __shared__ float red[256];
    int b = blockIdx.x;
    for (int j = threadIdx.x; j < CINN; j += blockDim.x)
        buf[j] = (j < EE) ? x_emb[((size_t)b * LLEN + t) * EE + j]
                          : rv[(size_t)b * (NRR * HD) + (j - EE)];
    ln_finish(buf, red, CINN, g, bv, inp + (size_t)b * CINN);
}

__global__ void preout_ln(const float* __restrict__ hn, const float* __restrict__ rv,
                          const float* __restrict__ g, const float* __restrict__ bv,
                          float* __restrict__ out)
{
    __shared__ float buf[TOTD];
    __shared__ float red[256];
    int b = blockIdx.x;
    for (int j = threadIdx.x; j < TOTD; j += blockDim.x)
        buf[j] = (j < HH) ? hn[(size_t)b * HH + j]
                          : rv[(size_t)b * (NRR * HD) + (j - HH)];
    ln_finish(buf, red, TOTD, g, bv, out + (size_t)b * TOTD);
}

// ---------------------------------------------------------------------------
// LSTM pointwise: PyTorch gate order i,f,g,o
// ---------------------------------------------------------------------------
__global__ void lstm_update(const float* __restrict__ gates,
                            float* __restrict__ h, float* __restrict__ c)
{
    int b = blockIdx.x;
    const float* gr = gates + (size_t)b * (4 * HH);
    for (int j = threadIdx.x; j < HH; j += blockDim.x) {
        float ig = sigm_(gr[j]);
        float fg = sigm_(gr[HH + j]);
        float gg = tanhf(gr[2 * HH + j]);
        float og = sigm_(gr[3 * HH + j]);
        float cn = fg * c[(size_t)b * HH + j] + ig * gg;
        c[(size_t)b * HH + j] = cn;
        h[(size_t)b * HH + j] = og * tanhf(cn);
    }
}

// out[b][n] = act(dot(X[b,:K], W[n,:K]) + bias[n]); act: 1 sigmoid (ws head, N=1)
__global__ void matvec_act(const float* __restrict__ X, const float* __restrict__ W,
                           const float* __restrict__ bias, float* __restrict__ out,
                           int N, int K, int act)
{
    int id = blockIdx.x * blockDim.x + threadIdx.x;
    if (id >= BB * N) return;
    int b = id / N, n = id % N;
    const float* xr = X + (size_t)b * K;
    const float* wr = W + (size_t)n * K;
    float s = bias ? bias[n] : 0.f;
    for (int k = 0; k < K; ++k) s += xr[k] * wr[k];
    if (act == 1) s = sigm_(s);
    else if (act == 2) s = tanhf(s);
    out[id] = s;
}

// ---------------------------------------------------------------------------
// 4-head attention over memory slots. grid = B*NHEAD, block = 128.
// qkv rows: [B*M, 384] with q|k|v each 128-wide, head h at cols h*32.
// q/k/v staged in LDS (48 KB), online softmax per row.
// ---------------------------------------------------------------------------
__global__ void attn_mem(const float* __restrict__ qkv, float* __restrict__ o)
{
    __shared__ float qs[MMEM][HDIM];
    __shared__ float ks[MMEM][HDIM];
    __shared__ float vs[MMEM][HDIM];
    int b = blockIdx.x / NHEAD, h = blockIdx.x % NHEAD;
    int base = b * MMEM;
    int cq = h * HDIM, ck = 128 + h * HDIM, cv = 256 + h * HDIM;
    for (int idx = threadIdx.x; idx < MMEM * HDIM; idx += blockDim.x) {
        int r = idx >> 5, cc = idx & 31;
        const float* row = qkv + (size_t)(base + r) * 384;
        qs[r][cc] = row[cq + cc];
        ks[r][cc] = row[ck + cc];
        vs[r][cc] = row[cv + cc];
    }
    __syncthreads();
    int r = threadIdx.x;  // one row per thread (128 threads)
    const float scale = 0.17677669529663687f;  // 1/sqrt(32)
    float m = -1e30f, ssum = 0.f;
    for (int c2 = 0; c2 < MMEM; ++c2) {
        float s = 0.f;
#pragma unroll
        for (int d = 0; d < HDIM; ++d) s += qs[r][d] * ks[c2][d];
        s *= scale;
        float mn = fmaxf(m, s);
        ssum = ssum * __expf(m - mn) + __expf(s - mn);
        m = mn;
    }
    float accv[HDIM];
#pragma unroll
    for (int d = 0; d < HDIM; ++d) accv[d] = 0.f;
    for (int c2 = 0; c2 < MMEM; ++c2) {
        float s = 0.f;
#pragma unroll
        for (int d = 0; d < HDIM; ++d) s += qs[r][d] * ks[c2][d];
        float w = __expf(s * scale - m) / ssum;
#pragma unroll
        for (int d = 0; d < HDIM; ++d) accv[d] += w * vs[c2][d];
    }
    float* orow = o + (size_t)(base + r) * HD + h * HDIM;
#pragma unroll
    for (int d = 0; d < HDIM; ++d) orow[d] = accv[d];
}

// ww[b,m] = softmax_m(wkn[b] . mn[b,m,:]) * ws[b].  grid=B, block=128
__global__ void write_weights(const float* __restrict__ wkn, const float* __restrict__ mn,
                              const float* __restrict__ wsg, float* __restrict__ ww)
{
    __shared__ float red[128];
    int b = blockIdx.x, m = threadIdx.x;
    const float* kr = wkn + (size_t)b * HD;
    const float* mr = mn + (size_t)(b * MMEM + m) * HD;
    float s = 0.f;
    for (int d = 0; d < HD; ++d) s += kr[d] * mr[d];
    red[m] = s; __syncthreads();
    for (int off = 64; off > 0; off >>= 1) {
        if (m < off) red[m] = fmaxf(red[m], red[m + off]);
        __syncthreads();
    }
    float mx = red[0]; __syncthreads();
    float e = __expf(s - mx);
    red[m] = e; __syncthreads();
    for (int off = 64; off > 0; off >>= 1) {
        if (m < off) red[m] += red[m + off];
        __syncthreads();
    }
    ww[b * MMEM + m] = (e / red[0]) * wsg[b];
}

// mem = mt*(1 - ww*ev) + ww*av
__global__ void mem_update(const float* __restrict__ mt, const float* __restrict__ ww,
                           const float* __restrict__ ev, const float* __restrict__ av,
                           float* __restrict__ mem)
{
    int idx = blockIdx.x * blockDim.x + threadIdx.x;
    if (idx >= BB * MMEM * HD) return;
    int hq = idx % HD;
    int bm = idx / HD;
    int b = bm / MMEM;
    float w = ww[bm];
    mem[idx] = mt[idx] * (1.f - w * ev[b * HD + hq]) + w * av[b * HD + hq];
}

// rv[b,n,:] = softmax_m(rkn[b,n].mn2[b,m]) @ mt2[b,m,:].  grid=B*NR, block=128
__global__ void read_attn(const float* __restrict__ rkn, const float* __restrict__ mn2,
                          const float* __restrict__ mt2, float* __restrict__ rv)
{
    __shared__ float red[128];
    __shared__ float wsh[128];
    int bn = blockIdx.x;
    int b = bn / NRR;
    int m = threadIdx.x;
    const float* kr = rkn + (size_t)bn * HD;
    const float* mr = mn2 + (size_t)(b * MMEM + m) * HD;
    float s = 0.f;
    for (int d = 0; d < HD; ++d) s += kr[d] * mr[d];
    red[m] = s; __syncthreads();
    for (int off = 64; off > 0; off >>= 1) {
        if (m < off) red[m] = fmaxf(red[m], red[m + off]);
        __syncthreads();
    }
    float mx = red[0]; __syncthreads();
    float e = __expf(s - mx);
    red[m] = e; __syncthreads();
    for (int off = 64; off > 0; off >>= 1) {
        if (m < off) red[m] += red[m + off];
        __syncthreads();
    }
    wsh[m] = e / red[0]; __syncthreads();
    int d = threadIdx.x;
    float acc = 0.f;
    for (int mm = 0; mm < MMEM; ++mm)
        acc += wsh[mm] * mt2[(size_t)(b * MMEM + mm) * HD + d];
    rv[(size_t)bn * HD + d] = acc;
}

// ---------------------------------------------------------------------------
extern "C" void kernel_launch(void* const* d_in, const int* in_sizes, int n_in,
                              void* d_out, int out_size, void* d_ws, size_t ws_size,
                              hipStream_t stream)
{
    if (n_in < 53) return;
    const float* X = (const float*)d_in[0];
    auto F = [&](int i) { return (const float*)d_in[i]; };
    // param indices (flattened setup_inputs order)
    const float *in_g = F(1), *in_b = F(2);
    const float *Wih = F(3), *Whh = F(4), *bih = F(5), *bhh = F(6);
    const float *cn_g = F(7), *cn_b = F(8);
    const float *rk_w = F(9), *rk_b = F(10), *wk_w = F(11), *wk_b = F(12);
    const float *ws_w = F(13), *ws_b = F(14), *er_w = F(15), *er_b = F(16);
    const float *ad_w = F(17), *ad_b = F(18);
    // mem_tf 19..30, read_tf 31..42
    const float *rkn_g = F(43), *rkn_b = F(44);
    const float *wkn_g = F(45), *wkn_b = F(46);
    const float *mn_g = F(47), *mn_b = F(48);
    const float *po_g = F(49), *po_b = F(50);
    const float *proj_w = F(51), *proj_b = F(52);

    float* ws = (float*)d_ws;
    size_t off = 0;
    auto alloc = [&](size_t n) { float* p = ws + off; off += n; return p; };
    // carried state (contiguous, zeroed below)
    float* h   = alloc((size_t)BB * HH);
    float* c   = alloc((size_t)BB * HH);
    float* rv  = alloc((size_t)BB * NRR * HD);
    float* mem = alloc((size_t)BB * MMEM * HD);
    size_t carried = off;
    // scratch
    float* inp   = alloc((size_t)BB * CINN);
    float* gates = alloc((size_t)BB * 4 * HH);
    float* hn    = alloc((size_t)BB * HH);
    float* rk    = alloc((size_t)BB * NRR * HD);
    float* wk    = alloc((size_t)BB * HD);
    float* wsg   = alloc(BB);
    float* ev    = alloc((size_t)BB * HD);
    float* av    = alloc((size_t)BB * HD);
    float* qkv   = alloc((size_t)BB * MMEM * 384);
    float* attno = alloc((size_t)BB * MMEM * HD);
    float* proj  = alloc((size_t)BB * MMEM * HD);
    float* x1    = alloc((size_t)BB * MMEM * HD);
    float* ff1   = alloc((size_t)BB * MMEM * 256);
    float* ff2   = alloc((size_t)BB * MMEM * HD);
    float* mt    = alloc((size_t)BB * MMEM * HD);
    float* mt2   = alloc((size_t)BB * MMEM * HD);
    float* mn    = alloc((size_t)BB * MMEM * HD);
    float* mn2   = alloc((size_t)BB * MMEM * HD);
    float* wkn   = alloc((size_t)BB * HD);
    float* ww    = alloc((size_t)BB * MMEM);
    float* qkvr  = alloc((size_t)64 * 384);
    float* projr = alloc((size_t)64 * HD);
    float* x1r   = alloc((size_t)64 * HD);
    float* ff1r  = alloc((size_t)64 * 256);
    float* ff2r  = alloc((size_t)64 * HD);
    float* rkt   = alloc((size_t)64 * HD);
    float* rkn   = alloc((size_t)64 * HD);
    float* outln = alloc((size_t)BB * TOTD);
    if (off * sizeof(float) > ws_size) return;

    hipMemsetAsync(ws, 0, carried * sizeof(float), stream);

    // C = act(A@W^T + bias); N must be a multiple of 64
    auto gemm = [&](const float* A, int lda, const float* W, int ldw,
                    const float* bias, float* C, int ldc,
                    int Mr, int N, int K, int act = 0) {
        int tiles = (Mr / 16) * (N / 64);
        int blocks = (tiles + 3) / 4;  // 4 waves (128 thr) per block
        gemm64_wmma<<<blocks, 128, 0, stream>>>(A, lda, W, ldw, bias, C, ldc,
                                                Mr, N, K, act,
                                                nullptr, 0, nullptr, 0, 0, nullptr);
    };
    // fused dual GEMM (same output tile): C = A@W^T + A2@W2^T + bias + bias2
    auto gemm2 = [&](const float* A, int lda, const float* W, int ldw,
                     const float* bias, const float* A2, int lda2,
                     const float* W2, int ldw2, int K2, const float* bias2,
                     float* C, int ldc, int Mr, int N, int K) {
        int tiles = (Mr / 16) * (N / 64);
        int blocks = (tiles + 3) / 4;
        gemm64_wmma<<<blocks, 128, 0, stream>>>(A, lda, W, ldw, bias, C, ldc,
                                                Mr, N, K, 0,
                                                A2, lda2, W2, ldw2, K2, bias2);
    };

    // full transformer block on [B*M, 128] with mem_tf params (19..30)
    auto tblock_mem = [&](const float* xin, float* out) {
        gemm(xin, HD, F(19), HD, F(20), qkv, 384, BB * MMEM, 384, HD);
        attn_mem<<<BB * NHEAD, 128, 0, stream>>>(qkv, attno);
        gemm(attno, HD, F(21), HD, F(22), proj, HD, BB * MMEM, HD, HD);
        row_ln<<<BB * MMEM, 256, 0, stream>>>(proj, xin, F(27), F(28), x1, HD);
        gemm(x1, HD, F(23), HD, F(24), ff1, 256, BB * MMEM, 256, HD, 1);
        gemm(ff1, 256, F(25), 256, F(26), ff2, HD, BB * MMEM, HD, 256);
        row_ln<<<BB * MMEM, 256, 0, stream>>>(x1, ff2, F(29), F(30), out, HD);
    };
    // seq-len-1 transformer block on [64, 128] with read_tf params (31..42):
    // softmax over one key is 1, so attention output == V (qkv cols 256..383)
    auto tblock_read = [&](const float* xin, float* out) {
        gemm(xin, HD, F(31), HD, F(32), qkvr, 384, 64, 384, HD);
        gemm(qkvr + 256, 384, F(33), HD, F(34), projr, HD, 64, HD, HD);
        row_ln<<<64, 256, 0, stream>>>(projr, xin, F(39), F(40), x1r, HD);
        gemm(x1r, HD, F(35), HD, F(36), ff1r, 256, 64, 256, HD, 1);
        gemm(ff1r, 256, F(37), 256, F(38), ff2r, HD, 64, HD, 256);
        row_ln<<<64, 256, 0, stream>>>(x1r, ff2r, F(41), F(42), out, HD);
    };

    float* out_f = (float*)d_out;
    for (int t = 0; t < LLEN; ++t) {
        input_ln<<<BB, 256, 0, stream>>>(X, t, rv, in_g, in_b, inp);
        // gates = inp@Wih^T + h@Whh^T + bih + bhh  (single fused WMMA kernel)
        gemm2(inp, CINN, Wih, CINN, bih, h, HH, Whh, HH, HH, bhh,
              gates, 4 * HH, BB, 4 * HH, CINN);
        lstm_update<<<BB, 256, 0, stream>>>(gates, h, c);
        row_ln<<<BB, 256, 0, stream>>>(h, nullptr, cn_g, cn_b, hn, HH);
        // controller heads on the matrix pipe (N = 512/128, K = 1024)
        gemm(hn, HH, rk_w, HH, rk_b, rk, 512, BB, 512, HH, 0);
        gemm(hn, HH, wk_w, HH, wk_b, wk, 128, BB, 128, HH, 0);
        gemm(hn, HH, er_w, HH, er_b, ev, 128, BB, 128, HH, 2);  // sigmoid
        gemm(hn, HH, ad_w, HH, ad_b, av, 128, BB, 128, HH, 3);  // tanh
        matvec_act<<<1, 256, 0, stream>>>(hn, ws_w, ws_b, wsg, 1, HH, 1);
        // write memory
        tblock_mem(mem, mt);
        row_ln<<<BB * MMEM, 256, 0, stream>>>(mt, nullptr, mn_g, mn_b, mn, HD);
        row_ln<<<BB, 256, 0, stream>>>(wk, nullptr, wkn_g, wkn_b, wkn, HD);
        write_weights<<<BB, 128, 0, stream>>>(wkn, mn, wsg, ww);
        mem_update<<<(BB * MMEM * HD + 255) / 256, 256, 0, stream>>>(mt, ww, ev, av, mem);
        // read memory
        tblock_mem(mem, mt2);
        tblock_read(rk, rkt);
        row_ln<<<BB * MMEM, 256, 0, stream>>>(mt2, nullptr, mn_g, mn_b, mn2, HD);
        row_ln<<<64, 256, 0, stream>>>(rkt, nullptr, rkn_g, rkn_b, rkn, HD);
        read_attn<<<BB * NRR, 128, 0, stream>>>(rkn, mn2, mt2, rv);
        // output head
        preout_ln<<<BB, 256, 0, stream>>>(hn, rv, po_g, po_b, outln);
        gemm(outln, TOTD, proj_w, TOTD, proj_b, out_f + (size_t)t * OO,
             LLEN * OO, BB, OO, TOTD);
    }

    size_t ys = (size_t)BB * LLEN * OO;
    hipMemcpyAsync(out_f + ys, mem, (size_t)BB * MMEM * HD * sizeof(float),
                   hipMemcpyDeviceToDevice, stream);
    hipMemcpyAsync(out_f + ys + (size_t)BB * MMEM * HD, h,
                   (size_t)BB * HH * sizeof(float), hipMemcpyDeviceToDevice, stream);
    hipMemcpyAsync(out_f + ys + (size_t)BB * MMEM * HD + (size_t)BB * HH, c,
                   (size_t)BB * HH * sizeof(float), hipMemcpyDeviceToDevice, stream);
}